// FastRCNN_49031346651237
// MI455X (gfx1250) — compile-verified
//
#include <hip/hip_runtime.h>
#include <hip/hip_bf16.h>

typedef float v2f __attribute__((ext_vector_type(2)));
typedef float v8f __attribute__((ext_vector_type(8)));

#define N_ROWS 512
#define NPROP  248
#define NGT    8
#define CFEAT  512
#define HF     25
#define WF     40
#define K1     4608
#define HID    1024
#define NCLS   20
#define NREG   80

// ---------------------------------------------------------------------------
// 1) props = concat(proposals, gt) ; IoU match ; labels ; box-encode targets
// ---------------------------------------------------------------------------
__global__ __launch_bounds__(256)
void match_kernel(const float* __restrict__ proposals, const float* __restrict__ gt_boxes,
                  const int* __restrict__ gt_labels, float* __restrict__ props,
                  int* __restrict__ labels, float* __restrict__ regt)
{
    int n = blockIdx.x * 256 + threadIdx.x;
    if (n >= N_ROWS) return;
    int b = n >> 8, r = n & 255;
    float p0, p1, p2, p3;
    if (r < NPROP) {
        const float* q = proposals + ((b * NPROP + r) << 2);
        p0 = q[0]; p1 = q[1]; p2 = q[2]; p3 = q[3];
    } else {
        const float* q = gt_boxes + ((b * NGT + (r - NPROP)) << 2);
        p0 = q[0]; p1 = q[1]; p2 = q[2]; p3 = q[3];
    }
    props[n * 4 + 0] = p0; props[n * 4 + 1] = p1;
    props[n * 4 + 2] = p2; props[n * 4 + 3] = p3;

    float area_p = (p2 - p0) * (p3 - p1);
    float best = -1.0f; int bg = 0;
    float g0b = 0.f, g1b = 0.f, g2b = 0.f, g3b = 0.f;
    for (int g = 0; g < NGT; ++g) {
        const float* q = gt_boxes + ((b * NGT + g) << 2);
        float g0 = q[0], g1 = q[1], g2 = q[2], g3 = q[3];
        float ag = (g2 - g0) * (g3 - g1);
        float lx = fmaxf(p0, g0), ly = fmaxf(p1, g1);
        float rx = fminf(p2, g2), ry = fminf(p3, g3);
        float iw = fmaxf(rx - lx, 0.f), ih = fmaxf(ry - ly, 0.f);
        float inter = iw * ih;
        float iou = inter / (area_p + ag - inter);
        if (iou > best) { best = iou; bg = g; g0b = g0; g1b = g1; g2b = g2; g3b = g3; }
    }
    labels[n] = (best >= 0.5f) ? gt_labels[b * NGT + bg] : 0;

    float pw = p2 - p0, ph = p3 - p1;
    float pcx = p0 + 0.5f * pw, pcy = p1 + 0.5f * ph;
    float gw = g2b - g0b, gh = g3b - g1b;
    float gcx = g0b + 0.5f * gw, gcy = g1b + 0.5f * gh;
    regt[n * 4 + 0] = (gcx - pcx) / pw;
    regt[n * 4 + 1] = (gcy - pcy) / ph;
    regt[n * 4 + 2] = logf(gw / pw);
    regt[n * 4 + 3] = logf(gh / ph);
}

// ---------------------------------------------------------------------------
// 2) RoIPool 3x3 (torchvision math): x[n, c*9 + ph*3 + pw]
// ---------------------------------------------------------------------------
__global__ __launch_bounds__(256)
void roipool_kernel(const float* __restrict__ feat, const float* __restrict__ props,
                    float* __restrict__ x)
{
    int n = blockIdx.x;
    int b = n >> 8;
    float p0 = props[n * 4 + 0], p1 = props[n * 4 + 1];
    float p2 = props[n * 4 + 2], p3 = props[n * 4 + 3];
    float x1 = rintf(p0 * 0.03125f), y1 = rintf(p1 * 0.03125f);
    float x2 = rintf(p2 * 0.03125f), y2 = rintf(p3 * 0.03125f);
    float bw = fmaxf(x2 - x1 + 1.f, 1.f) * (1.f / 3.f);
    float bh = fmaxf(y2 - y1 + 1.f, 1.f) * (1.f / 3.f);

    int wsi[3], wei[3], hsi[3], hei[3];
#pragma unroll
    for (int p = 0; p < 3; ++p) {
        float wsf = fminf(fmaxf(floorf((float)p * bw) + x1, 0.f), (float)WF);
        float wef = fminf(fmaxf(ceilf((float)(p + 1) * bw) + x1, 0.f), (float)WF);
        wsi[p] = (int)wsf; wei[p] = (int)wef;
        float hsf = fminf(fmaxf(floorf((float)p * bh) + y1, 0.f), (float)HF);
        float hef = fminf(fmaxf(ceilf((float)(p + 1) * bh) + y1, 0.f), (float)HF);
        hsi[p] = (int)hsf; hei[p] = (int)hef;
    }

    for (int c = threadIdx.x; c < CFEAT; c += blockDim.x) {
        const float* f = feat + (long)(b * CFEAT + c) * (HF * WF);
        float* xo = x + (long)n * K1 + c * 9;
#pragma unroll
        for (int ph = 0; ph < 3; ++ph) {
#pragma unroll
            for (int pw = 0; pw < 3; ++pw) {
                bool empty = (wei[pw] <= wsi[pw]) || (hei[ph] <= hsi[ph]);
                float v = -INFINITY;
                for (int h = hsi[ph]; h < hei[ph]; ++h)
                    for (int w = wsi[pw]; w < wei[pw]; ++w)
                        v = fmaxf(v, f[h * WF + w]);
                xo[ph * 3 + pw] = empty ? 0.f : v;
            }
        }
    }
}

// ---------------------------------------------------------------------------
// 3) FP32 WMMA GEMM with M-direction register blocking.
//    C[M,N] = relu?(A[M,K] @ B[K,N] + bias)
//    One wave owns a 64(M) x 16(N) tile = 4 accumulators; one B fragment
//    feeds 4 V_WMMA_F32_16X16X4_F32 ops. Compile-time K,N => immediate
//    offsets on global_load, no per-iteration 64-bit address math.
//    Block = 8 waves = 64(M) x 128(N); grid = (M/64, N/128).
// ---------------------------------------------------------------------------
template<int K, int N>
__global__ __launch_bounds__(256)
void gemm_relu_wmma(const float* __restrict__ A, const float* __restrict__ Bw,
                    const float* __restrict__ bias, float* __restrict__ C,
                    int do_relu)
{
    const int lane = threadIdx.x & 31;
    const int wave = threadIdx.x >> 5;
    const int m0   = blockIdx.x * 64;
    const int n0   = blockIdx.y * 128 + wave * 16;
    const int half = lane >> 4;        // 0: K={0,1}, 1: K={2,3}
    const int l15  = lane & 15;
    const int koff = half * 2;

    const float* a0  = A  + (m0 + l15) * K + koff;       // sub-tile r: +r*16*K
    const float* bp0 = Bw + koff * N + n0 + l15;

    v8f acc0 = {0.f, 0.f, 0.f, 0.f, 0.f, 0.f, 0.f, 0.f};
    v8f acc1 = acc0, acc2 = acc0, acc3 = acc0;

    for (int kk = 0; kk < K; kk += 16) {
        // prefetch next B k-panel (global_prefetch_b8); speculative, safe at tail
        __builtin_prefetch(bp0 + (kk + 16) * N, 0, 1);
#pragma unroll
        for (int ks = 0; ks < 16; ks += 4) {
            const int k = kk + ks;
            v2f b;
            const float* bp = bp0 + k * N;
            b.x = bp[0];
            b.y = bp[N];

            v2f a;
            a.x = a0[k];              a.y = a0[k + 1];
            acc0 = __builtin_amdgcn_wmma_f32_16x16x4_f32(false, a, false, b,
                                                         (short)0, acc0, false, false);
            a.x = a0[16 * K + k];     a.y = a0[16 * K + k + 1];
            acc1 = __builtin_amdgcn_wmma_f32_16x16x4_f32(false, a, false, b,
                                                         (short)0, acc1, false, false);
            a.x = a0[32 * K + k];     a.y = a0[32 * K + k + 1];
            acc2 = __builtin_amdgcn_wmma_f32_16x16x4_f32(false, a, false, b,
                                                         (short)0, acc2, false, false);
            a.x = a0[48 * K + k];     a.y = a0[48 * K + k + 1];
            acc3 = __builtin_amdgcn_wmma_f32_16x16x4_f32(false, a, false, b,
                                                         (short)0, acc3, false, false);
        }
    }

    const int col  = n0 + l15;
    const float bv = bias ? bias[col] : 0.f;
    v8f accs[4] = {acc0, acc1, acc2, acc3};
#pragma unroll
    for (int r = 0; r < 4; ++r) {
        const int rbase = m0 + r * 16 + half * 8;
#pragma unroll
        for (int i = 0; i < 8; ++i) {
            float v = accs[r][i] + bv;
            if (do_relu) v = fmaxf(v, 0.f);
            C[(rbase + i) * N + col] = v;
        }
    }
}

// ---------------------------------------------------------------------------
// 4) Fused heads: [512,1024] x {Wc[1024,20] | Wr[1024,80]} padded to N=112.
//    Per-lane constant masks keep EXEC all-ones at every WMMA.
// ---------------------------------------------------------------------------
__global__ __launch_bounds__(256)
void heads_wmma(const float* __restrict__ H, const float* __restrict__ Wc,
                const float* __restrict__ bc, const float* __restrict__ Wr,
                const float* __restrict__ br, float* __restrict__ logits,
                float* __restrict__ deltas)
{
    const int lane = threadIdx.x & 31;
    const int wave = threadIdx.x >> 5;
    const int gid  = blockIdx.x * 8 + wave;   // 0..223 = 32 x 7 tiles
    const int m0   = (gid / 7) * 16;
    const int n0   = (gid % 7) * 16;
    const int half = lane >> 4;
    const int l15  = lane & 15;
    const int koff = half * 2;
    const int j    = n0 + l15;                // fused output column 0..111

    const float* wptr; int wstride; float mask;
    if (j < NCLS)               { wptr = Wc + j;          wstride = NCLS; mask = 1.f; }
    else if (j < NCLS + NREG)   { wptr = Wr + (j - NCLS); wstride = NREG; mask = 1.f; }
    else                        { wptr = Wc;              wstride = NCLS; mask = 0.f; }

    const float* arow = H + (m0 + l15) * HID + koff;
    v8f acc = {0.f, 0.f, 0.f, 0.f, 0.f, 0.f, 0.f, 0.f};
#pragma unroll 4
    for (int k = 0; k < HID; k += 4) {
        v2f a, b;
        a.x = arow[k];
        a.y = arow[k + 1];
        b.x = mask * wptr[(k + koff) * wstride];
        b.y = mask * wptr[(k + koff + 1) * wstride];
        acc = __builtin_amdgcn_wmma_f32_16x16x4_f32(false, a, false, b,
                                                    (short)0, acc, false, false);
    }

    const int rbase = m0 + half * 8;
    if (j < NCLS) {
        float bb = bc[j];
#pragma unroll
        for (int i = 0; i < 8; ++i)
            logits[(rbase + i) * NCLS + j] = acc[i] + bb;
    } else if (j < NCLS + NREG) {
        float bb = br[j - NCLS];
#pragma unroll
        for (int i = 0; i < 8; ++i)
            deltas[(rbase + i) * NREG + (j - NCLS)] = acc[i] + bb;
    }
}

// ---------------------------------------------------------------------------
// 5) Losses: softmax-CE mean + smooth-L1 on positives
// ---------------------------------------------------------------------------
__global__ __launch_bounds__(256)
void loss_kernel(const float* __restrict__ logits, const float* __restrict__ deltas,
                 const int* __restrict__ labels, const float* __restrict__ regt,
                 float* __restrict__ out_losses)
{
    __shared__ float scls[256], sreg[256], spos[256];
    int tid = threadIdx.x;
    float cls = 0.f, reg = 0.f, pos = 0.f;
    for (int n = tid; n < N_ROWS; n += 256) {
        const float* lg = logits + (long)n * NCLS;
        float m = lg[0];
        for (int jj = 1; jj < NCLS; ++jj) m = fmaxf(m, lg[jj]);
        float s = 0.f;
        for (int jj = 0; jj < NCLS; ++jj) s += expf(lg[jj] - m);
        float lse = logf(s) + m;
        int lab = labels[n];
        cls += lse - lg[lab];
        if (lab > 0) {
            pos += 1.f;
            const float* dl = deltas + (long)n * NREG + lab * 4;
            for (int jj = 0; jj < 4; ++jj) {
                float d  = dl[jj] - regt[n * 4 + jj];
                float ad = fabsf(d);
                reg += (ad < 1.f) ? 0.5f * d * d : (ad - 0.5f);
            }
        }
    }
    scls[tid] = cls; sreg[tid] = reg; spos[tid] = pos;
    __syncthreads();
    for (int s2 = 128; s2 > 0; s2 >>= 1) {
        if (tid < s2) {
            scls[tid] += scls[tid + s2];
            sreg[tid] += sreg[tid + s2];
            spos[tid] += spos[tid + s2];
        }
        __syncthreads();
    }
    if (tid == 0) {
        out_losses[0] = scls[0] / (float)N_ROWS;
        float np = fmaxf(spos[0], 1.f);
        out_losses[1] = sreg[0] / (4.f * np);
    }
}

// ---------------------------------------------------------------------------
extern "C" void kernel_launch(void* const* d_in, const int* in_sizes, int n_in,
                              void* d_out, int out_size, void* d_ws, size_t ws_size,
                              hipStream_t stream)
{
    const float* feat      = (const float*)d_in[0];
    const float* proposals = (const float*)d_in[1];
    const float* gt_boxes  = (const float*)d_in[2];
    const int*   gt_labels = (const int*)d_in[3];
    const float* W1 = (const float*)d_in[4];
    const float* b1 = (const float*)d_in[5];
    const float* W2 = (const float*)d_in[6];
    const float* b2 = (const float*)d_in[7];
    const float* Wc = (const float*)d_in[8];
    const float* bc = (const float*)d_in[9];
    const float* Wr = (const float*)d_in[10];
    const float* br = (const float*)d_in[11];

    char* ws = (char*)d_ws;
    float* props  = (float*)(ws + 0);                         //  8 KB
    int*   labels = (int*)  (ws + 8192);                      //  2 KB
    float* regt   = (float*)(ws + 10240);                     //  8 KB
    float* x      = (float*)(ws + 18432);                     //  9.4 MB
    float* h1     = (float*)(ws + 18432 + (size_t)N_ROWS * K1 * 4);
    float* h2     = h1 + (size_t)N_ROWS * HID;

    float* out    = (float*)d_out;
    float* logits = out;
    float* deltas = out + N_ROWS * NCLS;
    float* losses = out + N_ROWS * (NCLS + NREG);

    match_kernel  <<<2, 256, 0, stream>>>(proposals, gt_boxes, gt_labels, props, labels, regt);
    roipool_kernel<<<N_ROWS, 256, 0, stream>>>(feat, props, x);
    gemm_relu_wmma<K1,  HID><<<dim3(N_ROWS / 64, HID / 128), 256, 0, stream>>>(x,  W1, b1, h1, 1);
    gemm_relu_wmma<HID, HID><<<dim3(N_ROWS / 64, HID / 128), 256, 0, stream>>>(h1, W2, b2, h2, 1);
    heads_wmma    <<<28, 256, 0, stream>>>(h2, Wc, bc, Wr, br, logits, deltas);
    loss_kernel   <<<1, 256, 0, stream>>>(logits, deltas, labels, regt, losses);
}